// Sae_2826088481314
// MI455X (gfx1250) — compile-verified
//
#include <hip/hip_runtime.h>
#include <cstddef>

typedef __attribute__((ext_vector_type(2))) float v2f;
typedef __attribute__((ext_vector_type(8))) float v8f;

#define D_IN      512
#define N_LAT     16384
#define NTOK      4096
#define TOPK      32
#define M_PER_WG  64
#define NSPLIT    4
#define SLICE     (N_LAT / NSPLIT)   // 4096
#define CHUNK     128                // 8 waves x 16 latents
#define XS_STRIDE 514                // padded (even -> b64-aligned, low conflict)
#define PS_STRIDE 66
#define LSTRIDE   33

// d_out layout (floats): sae_out[2097152] | top_acts[131072] | top_idx[131072] | fvu | aux | multi
#define ACT_OFF   2097152
#define IDX_OFF   2228224
#define FVU_OFF   2359296
// partial top-k lists staged inside the (not-yet-written) sae_out region
#define PV_OFF    0
#define PI_OFF    524288
// ws layout (floats): [0]=l2, [1..512]=sum_x per channel, [513..1024]=sum_x2 per channel
#define WS_FLOATS 1025

__device__ __forceinline__ void topk_insert(float v, int n, float* lv, int* li,
                                            float& tmin, int& tpos) {
  if (v > tmin) {
    lv[tpos] = v; li[tpos] = n;
    float m = lv[0]; int mp = 0;
    #pragma unroll
    for (int e = 1; e < TOPK; ++e) {
      float u = lv[e];
      if (u < m) { m = u; mp = e; }
    }
    tmin = m; tpos = mp;
  }
}

__device__ __forceinline__ void store_relu_tile(float* pr, v8f c, float bias) {
  #pragma unroll
  for (int r = 0; r < 8; ++r) pr[r] = fmaxf(c[r] + bias, 0.0f);
}

// ---------------- Kernel 1: fused encoder GEMM (fp32 WMMA) + partial top-k ----
__global__ __launch_bounds__(256, 1)
void k_encode_topk(const float* __restrict__ x, const float* __restrict__ enc_w,
                   const float* __restrict__ enc_b, float* __restrict__ out) {
  extern __shared__ float smem[];
  float* x_s   = smem;                               // 64 x 514
  float* pre_s = x_s + M_PER_WG * XS_STRIDE;         // 128 x 66
  float* lvb   = pre_s + CHUNK * PS_STRIDE;          // 256 x 33 (vals)
  int*   lib   = (int*)(lvb + 256 * LSTRIDE);        // 256 x 33 (idx)

  const int tid  = threadIdx.x;
  const int lane = tid & 31;
  const int wave = tid >> 5;
  const int hi   = lane >> 4;   // 0/1: K-halves in A/B, M-halves in C
  const int ln   = lane & 15;

  const int t0    = blockIdx.x * M_PER_WG;   // first token of tile (single batch b)
  const int bb    = t0 >> 8;
  const int p0    = t0 & 255;
  const int slice = blockIdx.y;

  // stage x tile: x[b][c][p] -> x_s[token_local][c]  (coalesced global reads)
  const float* xb = x + (size_t)bb * (D_IN * 256) + p0;
  #pragma unroll
  for (int i = 0; i < (M_PER_WG * D_IN) / 256; ++i) {
    int e  = tid + i * 256;
    int c  = e >> 6;
    int tl = e & 63;
    x_s[tl * XS_STRIDE + c] = xb[c * 256 + tl];
  }

  float* mylv = lvb + tid * LSTRIDE;
  int*   myli = lib + tid * LSTRIDE;
  float tmin = -3.0e38f; int tpos = 0;
  #pragma unroll
  for (int e = 0; e < TOPK; ++e) { mylv[e] = -3.0e38f; myli[e] = 0; }
  __syncthreads();

  const int tok = tid & 63;     // token this thread tracks in top-k phase
  const int sub = tid >> 6;     // which 32-latent stripe of each chunk
  const float* aptr = x_s + ln * XS_STRIDE + 2 * hi;   // A operand base (per lane)

  for (int chunk = 0; chunk < SLICE / CHUNK; ++chunk) {
    const int nb = slice * SLICE + chunk * CHUNK + wave * 16;  // this wave's 16 latents
    const float* bptr = enc_w + (size_t)(nb + ln) * D_IN + 2 * hi;
    v8f c0 = {}, c1 = {}, c2 = {}, c3 = {};
    #pragma unroll 4
    for (int k0 = 0; k0 < D_IN; k0 += 4) {
      v2f bv = *(const v2f*)(bptr + k0);                         // enc_w (K x N tile)
      v2f a0 = *(const v2f*)(aptr + k0);                         // tokens 0..15
      v2f a1 = *(const v2f*)(aptr + 16 * XS_STRIDE + k0);        // tokens 16..31
      v2f a2 = *(const v2f*)(aptr + 32 * XS_STRIDE + k0);        // tokens 32..47
      v2f a3 = *(const v2f*)(aptr + 48 * XS_STRIDE + k0);        // tokens 48..63
      c0 = __builtin_amdgcn_wmma_f32_16x16x4_f32(false, a0, false, bv, (short)0, c0, false, false);
      c1 = __builtin_amdgcn_wmma_f32_16x16x4_f32(false, a1, false, bv, (short)0, c1, false, false);
      c2 = __builtin_amdgcn_wmma_f32_16x16x4_f32(false, a2, false, bv, (short)0, c2, false, false);
      c3 = __builtin_amdgcn_wmma_f32_16x16x4_f32(false, a3, false, bv, (short)0, c3, false, false);
    }
    const float bias = enc_b[nb + ln];
    // C layout: lane holds column n=ln, rows hi*8 + r  -> pre_s[latent_local][token]
    float* pr = pre_s + (wave * 16 + ln) * PS_STRIDE + hi * 8;
    store_relu_tile(pr + 0,  c0, bias);
    store_relu_tile(pr + 16, c1, bias);
    store_relu_tile(pr + 32, c2, bias);
    store_relu_tile(pr + 48, c3, bias);
    __syncthreads();

    // top-k update: 4 threads per token, 32 candidates each
    #pragma unroll 4
    for (int j = 0; j < 32; ++j) {
      int ll = sub * 32 + j;
      float v = pre_s[ll * PS_STRIDE + tok];
      topk_insert(v, slice * SLICE + chunk * CHUNK + ll, mylv, myli, tmin, tpos);
    }
    __syncthreads();
  }

  // intra-WG merge: thread t (<64) folds lists of threads t+64,t+128,t+192 into its own
  if (tid < M_PER_WG) {
    #pragma unroll
    for (int s = 1; s < 4; ++s) {
      float* ov = lvb + (tid + s * 64) * LSTRIDE;
      int*   oi = lib + (tid + s * 64) * LSTRIDE;
      for (int e = 0; e < TOPK; ++e)
        topk_insert(ov[e], oi[e], mylv, myli, tmin, tpos);
    }
    const int row = (t0 + tid) * NSPLIT + slice;
    float* pv = out + PV_OFF + (size_t)row * TOPK;
    int*   pi = (int*)out + PI_OFF + (size_t)row * TOPK;
    for (int e = 0; e < TOPK; ++e) { pv[e] = mylv[e]; pi[e] = myli[e]; }
  }
}

// ---------------- Kernel 2: merge 4 slice-lists per token, sort desc, emit ----
__global__ __launch_bounds__(64)
void k_merge_sort(float* __restrict__ out) {
  __shared__ float mv[64 * LSTRIDE];
  __shared__ int   mi[64 * LSTRIDE];
  const int T = blockIdx.x * 64 + threadIdx.x;
  const float* pv = out + PV_OFF;
  const int*   pi = (const int*)out + PI_OFF;
  float* lv = mv + threadIdx.x * LSTRIDE;
  int*   li = mi + threadIdx.x * LSTRIDE;
  float tmin = 3.0e38f; int tpos = 0;
  for (int e = 0; e < TOPK; ++e) {
    size_t o = (size_t)(T * NSPLIT) * TOPK + e;
    float v = pv[o];
    lv[e] = v; li[e] = pi[o];
    if (v < tmin) { tmin = v; tpos = e; }
  }
  for (int s = 1; s < NSPLIT; ++s) {
    for (int e = 0; e < TOPK; ++e) {
      size_t o = (size_t)(T * NSPLIT + s) * TOPK + e;
      topk_insert(pv[o], pi[o], lv, li, tmin, tpos);
    }
  }
  for (int i = 0; i < TOPK - 1; ++i) {     // selection sort, descending
    float best = lv[i]; int bj = i;
    for (int j = i + 1; j < TOPK; ++j)
      if (lv[j] > best) { best = lv[j]; bj = j; }
    float tv = lv[i]; lv[i] = lv[bj]; lv[bj] = tv;
    int   ti = li[i]; li[i] = li[bj]; li[bj] = ti;
  }
  const int bb = T >> 8, p = T & 255;      // [B,k,H,W] layout
  for (int e = 0; e < TOPK; ++e) {
    out[ACT_OFF + (size_t)bb * 8192 + e * 256 + p] = lv[e];
    out[IDX_OFF + (size_t)bb * 8192 + e * 256 + p] = (float)li[e];
  }
}

// ---------------- Kernel 3: sparse decode + l2 loss ---------------------------
__global__ __launch_bounds__(256)
void k_decode(const float* __restrict__ x, const float* __restrict__ dec_w,
              const float* __restrict__ dec_b, float* __restrict__ out,
              float* __restrict__ ws) {
  __shared__ float a_s[16][TOPK];
  __shared__ int   i_s[16][TOPK];
  __shared__ float red[8];
  const int tid  = threadIdx.x;
  const int tok0 = blockIdx.x * 16;
  for (int e = tid; e < 16 * TOPK; e += 256) {
    int t = e >> 5, j = e & 31;
    int token = tok0 + t;
    int bb = token >> 8, p = token & 255;
    a_s[t][j] = out[ACT_OFF + (size_t)bb * 8192 + j * 256 + p];
    i_s[t][j] = (int)out[IDX_OFF + (size_t)bb * 8192 + j * 256 + p];
  }
  __syncthreads();
  const int t  = tid & 15;
  const int cb = (tid >> 4) * 32;
  const int token = tok0 + t;
  const int bb = token >> 8, p = token & 255;
  float av[TOPK]; int iv[TOPK];
  #pragma unroll
  for (int e = 0; e < TOPK; ++e) { av[e] = a_s[t][e]; iv[e] = i_s[t][e]; }
  const float* xp = x   + (size_t)bb * 131072 + p;
  float*       op = out + (size_t)bb * 131072 + p;
  float l2 = 0.0f;
  #pragma unroll 2
  for (int j = 0; j < 32; ++j) {
    int c = cb + j;
    const float* dw = dec_w + (size_t)c * N_LAT;
    float acc = dec_b[c];
    #pragma unroll
    for (int e = 0; e < TOPK; ++e) acc += av[e] * dw[iv[e]];
    op[(size_t)c * 256] = acc;
    float err = acc - xp[(size_t)c * 256];
    l2 += err * err;
  }
  #pragma unroll
  for (int off = 16; off > 0; off >>= 1) l2 += __shfl_down(l2, off, 32);
  if ((tid & 31) == 0) red[tid >> 5] = l2;
  __syncthreads();
  if (tid == 0) {
    float s = 0.0f;
    #pragma unroll
    for (int i = 0; i < 8; ++i) s += red[i];
    atomicAdd(ws, s);
  }
}

// ---------------- Kernel 4: per-channel x statistics --------------------------
__global__ __launch_bounds__(256)
void k_stats(const float* __restrict__ x, float* __restrict__ ws) {
  __shared__ float rs[8], rq[8];
  const int tid = threadIdx.x;
  const int bc  = blockIdx.x;          // b*512 + c
  const int c   = bc & 511;
  float v = x[(size_t)bc * 256 + tid];
  float s = v, q = v * v;
  #pragma unroll
  for (int off = 16; off > 0; off >>= 1) {
    s += __shfl_down(s, off, 32);
    q += __shfl_down(q, off, 32);
  }
  if ((tid & 31) == 0) { rs[tid >> 5] = s; rq[tid >> 5] = q; }
  __syncthreads();
  if (tid == 0) {
    float ss = 0.0f, qq = 0.0f;
    #pragma unroll
    for (int i = 0; i < 8; ++i) { ss += rs[i]; qq += rq[i]; }
    atomicAdd(ws + 1 + c, ss);
    atomicAdd(ws + 1 + 512 + c, qq);
  }
}

// ---------------- Kernel 5: fvu + zero scalars --------------------------------
__global__ __launch_bounds__(256)
void k_fvu(const float* __restrict__ ws, float* __restrict__ out) {
  __shared__ float red[8];
  const int tid = threadIdx.x;
  float tv = 0.0f;
  for (int c = tid; c < 512; c += 256) {
    float S = ws[1 + c], Q = ws[1 + 512 + c];
    tv += Q - S * S * (1.0f / 4096.0f);
  }
  #pragma unroll
  for (int off = 16; off > 0; off >>= 1) tv += __shfl_down(tv, off, 32);
  if ((tid & 31) == 0) red[tid >> 5] = tv;
  __syncthreads();
  if (tid == 0) {
    float t = 0.0f;
    #pragma unroll
    for (int i = 0; i < 8; ++i) t += red[i];
    out[FVU_OFF]     = ws[0] / t;
    out[FVU_OFF + 1] = 0.0f;
    out[FVU_OFF + 2] = 0.0f;
  }
}

__global__ void k_zero(float* __restrict__ ws) {
  int i = blockIdx.x * 256 + threadIdx.x;
  if (i < WS_FLOATS) ws[i] = 0.0f;
}

extern "C" void kernel_launch(void* const* d_in, const int* in_sizes, int n_in,
                              void* d_out, int out_size, void* d_ws, size_t ws_size,
                              hipStream_t stream) {
  (void)in_sizes; (void)n_in; (void)out_size; (void)ws_size;
  const float* x     = (const float*)d_in[0];
  const float* enc_w = (const float*)d_in[1];
  const float* enc_b = (const float*)d_in[2];
  const float* dec_w = (const float*)d_in[3];
  const float* dec_b = (const float*)d_in[4];
  float* out = (float*)d_out;
  float* ws  = (float*)d_ws;

  const size_t smem1 =
      (size_t)(M_PER_WG * XS_STRIDE + CHUNK * PS_STRIDE + 256 * LSTRIDE) * sizeof(float)
      + (size_t)256 * LSTRIDE * sizeof(int);   // = 232,960 B (< 320 KB WGP LDS)
  (void)hipFuncSetAttribute(reinterpret_cast<const void*>(k_encode_topk),
                            hipFuncAttributeMaxDynamicSharedMemorySize, (int)smem1);

  k_zero<<<(WS_FLOATS + 255) / 256, 256, 0, stream>>>(ws);
  k_encode_topk<<<dim3(NTOK / M_PER_WG, NSPLIT), 256, smem1, stream>>>(x, enc_w, enc_b, out);
  k_merge_sort<<<NTOK / 64, 64, 0, stream>>>(out);
  k_decode<<<NTOK / 16, 256, 0, stream>>>(x, dec_w, dec_b, out, ws);
  k_stats<<<16 * 512, 256, 0, stream>>>(x, ws);
  k_fvu<<<1, 256, 0, stream>>>(ws, out);
}